// VanillaRNN_84499186581932
// MI455X (gfx1250) — compile-verified
//
#include <hip/hip_runtime.h>
#include <math.h>

typedef __attribute__((ext_vector_type(2))) float v2f;
typedef __attribute__((ext_vector_type(4))) float v4f;
typedef __attribute__((ext_vector_type(8))) float v8f;

#if defined(__has_builtin)
# if __has_builtin(__builtin_amdgcn_tanhf)
#  define FAST_TANH(v) __builtin_amdgcn_tanhf(v)
# else
#  define FAST_TANH(v) tanhf(v)
# endif
#else
# define FAST_TANH(v) tanhf(v)
#endif

#define T_LEN   1024
#define I_DIM   3
#define H_DIM   32
#define LSTRIDE 36   // padded LDS row stride (floats): conflict-free b64 A-loads, 16B-aligned b128
#define PF_DIST 48   // x prefetch distance in timesteps (576 B, ~4.5 128B lines ahead)

// Two waves per block cooperate on 16 batch rows (512 waves total):
//   wave w computes the 16x16 N-tile (hidden cols 16w..16w+15) with 8 x
//   V_WMMA_F32_16X16X4_F32 (two 4-deep chains), v_tanh, writes its half of h
//   to a double-buffered LDS tile, one s_barrier, then both waves reload the
//   full h in A-layout and emit 128B-line non-temporal stores of out[b,t,:].
__global__ __launch_bounds__(64)
void rnn_fused(const float* __restrict__ x,
               const float* __restrict__ W_ih,
               const float* __restrict__ b_ih,
               const float* __restrict__ W_hh,
               const float* __restrict__ b_hh,
               const float* __restrict__ W_fc,
               const float* __restrict__ b_fc,
               float* __restrict__ out,
               float* __restrict__ hn)
{
  __shared__ float lds[2][16 * LSTRIDE];   // double-buffered h tile [m][n]
  const int lane = threadIdx.x & 31;
  const int w    = threadIdx.x >> 5;       // N-half owned by this wave (0 or 1)
  const int b0   = blockIdx.x * 16;        // 16 batch rows per block
  const int half = lane >> 4;              // lane half: K-pair select (A/B), M-half (C/D)
  const int l16  = lane & 15;

  // ---- W_hh: only this wave's N-half, resident as WMMA B operands (16 VGPRs) ----
  const int np = l16 + 16 * w;             // hidden output column n'
  v2f Bm[8];
#pragma unroll
  for (int s = 0; s < 8; ++s)
    Bm[s] = *(const v2f*)(W_hh + np * H_DIM + 4 * s + 2 * half);

  // ---- input-projection constants for column n' ----
  const float wih0 = W_ih[np * I_DIM + 0];
  const float wih1 = W_ih[np * I_DIM + 1];
  const float wih2 = W_ih[np * I_DIM + 2];
  const float bias = b_ih[np] + b_hh[np];

  // x base pointers: VGPR r carries batch row (r + 8*half), broadcast across 16 lanes
  const float* xb[8];
#pragma unroll
  for (int r = 0; r < 8; ++r)
    xb[r] = x + (size_t)(b0 + r + 8 * half) * T_LEN * I_DIM;

  // h = 0 in A-layout (shared K-slices, 16 VGPRs)
  v2f A[8];
#pragma unroll
  for (int s = 0; s < 8; ++s) A[s] = (v2f){0.0f, 0.0f};

  const int m8 = lane >> 3;        // out-store row within 4-row chunk
  const int n0 = (lane & 7) * 4;   // out-store float4 column

  for (int t = 0; t < T_LEN; ++t) {
    // ---- prefetch x ~48 steps ahead (uniform branch, every 8th step, clamped) ----
    if ((t & 7) == 0) {
      const int tp = t + PF_DIST;
      if (tp < T_LEN) {
#pragma unroll
        for (int r = 0; r < 8; ++r)
          __builtin_prefetch(xb[r] + I_DIM * tp, 0, 3);
      }
    }

    // ---- broadcast x loads (2 rows x 12B per load, shared by half-waves) ----
    float x0[8], x1[8], x2[8];
#pragma unroll
    for (int r = 0; r < 8; ++r) {
      const float* p = xb[r] + I_DIM * t;
      x0[r] = p[0]; x1[r] = p[1]; x2[r] = p[2];
    }

    // ---- C = x @ W_ih^T + b_ih + b_hh in accumulator layout; second chain from 0 ----
    v8f accA, accB;
#pragma unroll
    for (int r = 0; r < 8; ++r) {
      accA[r] = fmaf(x0[r], wih0, fmaf(x1[r], wih1, fmaf(x2[r], wih2, bias)));
      accB[r] = 0.0f;
    }

    // ---- D += h x W_hh^T (this N-half): two interleaved 4-deep WMMA chains ----
#pragma unroll
    for (int s = 0; s < 4; ++s) {
      accA = __builtin_amdgcn_wmma_f32_16x16x4_f32(false, A[2 * s],     false, Bm[2 * s],
                                                   (short)0, accA, false, false);
      accB = __builtin_amdgcn_wmma_f32_16x16x4_f32(false, A[2 * s + 1], false, Bm[2 * s + 1],
                                                   (short)0, accB, false, false);
    }

    // ---- merge chains + activation ----
#pragma unroll
    for (int r = 0; r < 8; ++r) accA[r] = FAST_TANH(accA[r] + accB[r]);

    // ---- write our N-half into the ping-pong LDS tile [m][n] ----
    float* lb = lds[t & 1];
#pragma unroll
    for (int r = 0; r < 8; ++r)
      lb[(r + 8 * half) * LSTRIDE + np] = accA[r];

    __syncthreads();   // pair exchange: both halves of h now visible

    // ---- reload full h as A-layout K-slices (ds_load_b64, bank-clean) ----
#pragma unroll
    for (int s = 0; s < 8; ++s)
      A[s] = *(const v2f*)(lb + l16 * LSTRIDE + 4 * s + 2 * half);

    // ---- out[b, t, :]: wave w stores rows 8w..8w+7 as full 128B lines, NT ----
#pragma unroll
    for (int cc = 0; cc < 2; ++cc) {
      const int m = 8 * w + 4 * cc + m8;
      v4f v = *(const v4f*)(lb + m * LSTRIDE + n0);
      __builtin_nontemporal_store(
          v, (v4f*)(out + (((size_t)(b0 + m) * T_LEN + t) * H_DIM + n0)));
    }
  }

  // ---- fc head (wave 0): lane (l16, half) -> h_n[b0+l16, half] ----
  if (w == 0) {
    const float* lb = lds[(T_LEN - 1) & 1];
    float s = b_fc[half];
#pragma unroll
    for (int k = 0; k < H_DIM; ++k)
      s = fmaf(lb[l16 * LSTRIDE + k], W_fc[half * H_DIM + k], s);
    hn[(size_t)(b0 + l16) * 2 + half] = s;
  }
}

extern "C" void kernel_launch(void* const* d_in, const int* in_sizes, int n_in,
                              void* d_out, int out_size, void* d_ws, size_t ws_size,
                              hipStream_t stream) {
  const float* x    = (const float*)d_in[0];
  const float* W_ih = (const float*)d_in[1];
  const float* b_ih = (const float*)d_in[2];
  const float* W_hh = (const float*)d_in[3];
  const float* b_hh = (const float*)d_in[4];
  const float* W_fc = (const float*)d_in[5];
  const float* b_fc = (const float*)d_in[6];

  float* out = (float*)d_out;
  const int B = in_sizes[0] / (T_LEN * I_DIM);           // 4096
  float* hn = out + (size_t)B * T_LEN * H_DIM;           // h_n follows `out` flat

  const int blocks = B / 16;                             // 2 waves/block, 16 rows/block
  rnn_fused<<<blocks, 64, 0, stream>>>(x, W_ih, b_ih, W_hh, b_hh, W_fc, b_fc, out, hn);
}